// BioOTSRSolver_60447369723975
// MI455X (gfx1250) — compile-verified
//
#include <hip/hip_runtime.h>
#include <stdint.h>

// ---------------------------------------------------------------------------
// BioOTSRSolver for MI455X (gfx1250, wave32).
// Memory-bound elementwise kernel (no GEMM structure -> WMMA not applicable).
// Strategy: async-DMA inputs global->LDS (coalesced b128, padded LDS layout),
// compute per-element in registers, stage outputs in LDS, async-DMA LDS->global.
// Roofline: ~178 MB traffic / 23.3 TB/s ~= 7.6 us floor; memory-bound.
// Full blocks (the only kind for B=262144) take a branch-free staging path.
// ---------------------------------------------------------------------------

#define TILE 256            // elements per block == threads per block (8 waves)
#define KP_STRIDE 76        // dwords/element in LDS: 72 data + 4 pad (16B aligned, <=2-way)
#define OR_STRIDE 44        // 36 data + 8 pad
#define SC_STRIDE 20        // 16 data + 4 pad
// LDS: 256*(76+44+20)*4 = 143360 B; output staging (256*46*4 = 47104 B) reuses kp area.

#if defined(__has_builtin)
#if __has_builtin(__builtin_amdgcn_global_load_async_to_lds_b128)
#define HAVE_ASYNC_LD 1
#endif
#if __has_builtin(__builtin_amdgcn_global_store_async_from_lds_b128)
#define HAVE_ASYNC_ST 1
#endif
#endif
#ifndef HAVE_ASYNC_LD
#define HAVE_ASYNC_LD 0
#endif
#ifndef HAVE_ASYNC_ST
#define HAVE_ASYNC_ST 0
#endif

typedef int v4i __attribute__((ext_vector_type(4)));
typedef __attribute__((address_space(1))) v4i gv4i;   // global int4
typedef __attribute__((address_space(3))) v4i lv4i;   // LDS int4

__device__ __forceinline__ lv4i* to_lds(const void* p) {
  // Generic LDS aperture address: low 32 bits are the LDS byte offset.
  return (lv4i*)(uint32_t)(uintptr_t)p;
}
__device__ __forceinline__ gv4i* to_glb(const void* p) {
  return (gv4i*)(uintptr_t)p;
}

__device__ __forceinline__ void async_g2l_b128(const void* g, void* l) {
#if HAVE_ASYNC_LD
  __builtin_amdgcn_global_load_async_to_lds_b128(to_glb(g), to_lds(l), 0, 0);
#else
  *(float4*)l = *(const float4*)g;   // coalesced b128 load + ds_store_b128
#endif
}
__device__ __forceinline__ void async_l2g_b128(void* g, const void* l) {
#if HAVE_ASYNC_ST
  __builtin_amdgcn_global_store_async_from_lds_b128(to_glb(g), to_lds(l), 0, 0);
#else
  *(float4*)g = *(const float4*)l;
#endif
}
__device__ __forceinline__ void wait_async_zero() {
#if HAVE_ASYNC_LD || HAVE_ASYNC_ST
#if defined(__has_builtin) && __has_builtin(__builtin_amdgcn_s_wait_asynccnt)
  __builtin_amdgcn_s_wait_asynccnt(0);
#else
  asm volatile("s_wait_asynccnt 0x0" ::: "memory");
#endif
#endif
}

__global__ void __launch_bounds__(TILE)
bio_otsr_kernel(const float* __restrict__ kp3d,
                const float* __restrict__ ortho,
                const float* __restrict__ scalar,
                float* __restrict__ out, int B) {
  __shared__ float s_kp[TILE * KP_STRIDE];   // 77824 B (reused for output staging)
  __shared__ float s_or[TILE * OR_STRIDE];   // 45056 B
  __shared__ float s_sc[TILE * SC_STRIDE];   // 20480 B

  const int t    = threadIdx.x;
  const int base = blockIdx.x * TILE;
  const int nb   = (B - base < TILE) ? (B - base) : TILE;

  // ---- Stage inputs: global side linear in chunk index c (offset = c*16B),
  //      LDS side scattered into padded per-element strides. ----
  const char* gk = (const char*)(kp3d + (size_t)base * 72);
  const char* go = (const char*)(ortho + (size_t)base * 36);
  const char* gs = (const char*)(scalar + (size_t)base * 16);
  char* lk = (char*)s_kp;
  char* lo = (char*)s_or;
  char* ls = (char*)s_sc;

  if (nb == TILE) {
    // Branch-free fast path: constant trip counts, no exec-mask churn.
#pragma unroll
    for (int it = 0; it < 18; ++it) {           // kp3d: 18 b128 chunks / element
      int c = t + it * TILE;
      async_g2l_b128(gk + (size_t)c * 16, lk + (size_t)(c + c / 18) * 16);
    }
#pragma unroll
    for (int it = 0; it < 9; ++it) {            // ortho: 9 chunks / element
      int c = t + it * TILE;
      async_g2l_b128(go + (size_t)c * 16, lo + (size_t)(c + 2 * (c / 9)) * 16);
    }
#pragma unroll
    for (int it = 0; it < 4; ++it) {            // scalar: 4 chunks / element
      int c = t + it * TILE;
      async_g2l_b128(gs + (size_t)c * 16, ls + (size_t)(c + c / 4) * 16);
    }
  } else {
    const int nck = nb * 18, nco = nb * 9, ncs = nb * 4;
#pragma unroll
    for (int it = 0; it < 18; ++it) {
      int c = t + it * TILE;
      if (c < nck) async_g2l_b128(gk + (size_t)c * 16, lk + (size_t)(c + c / 18) * 16);
    }
#pragma unroll
    for (int it = 0; it < 9; ++it) {
      int c = t + it * TILE;
      if (c < nco) async_g2l_b128(go + (size_t)c * 16, lo + (size_t)(c + 2 * (c / 9)) * 16);
    }
#pragma unroll
    for (int it = 0; it < 4; ++it) {
      int c = t + it * TILE;
      if (c < ncs) async_g2l_b128(gs + (size_t)c * 16, ls + (size_t)(c + c / 4) * 16);
    }
  }
  wait_async_zero();
  __syncthreads();

  // ---- Compute one element per thread, fully in registers. ----
  float res[46];
  if (t < nb) {
    const float* kp = s_kp + t * KP_STRIDE;
    const float* ov = s_or + t * OR_STRIDE;
    const float* sc = s_sc + t * SC_STRIDE;

#pragma unroll
    for (int k = 0; k < 8; ++k) res[k] = sc[k];    // thetas[D_IDX] = scalar[0:8]

    const int CH[10] = {1, 2, 4, 5, 7, 8, 16, 17, 18, 19};
    const int PA[10] = {0, 0, 1, 2, 4, 5, 13, 14, 16, 17};
#pragma unroll
    for (int i = 0; i < 10; ++i) {
      const int c3 = CH[i] * 3, p3 = PA[i] * 3;
      float dx = kp[c3 + 0] - kp[p3 + 0];
      float dy = kp[c3 + 1] - kp[p3 + 1];
      float dz = kp[c3 + 2] - kp[p3 + 2];
      float inv = 1.0f / fmaxf(sqrtf(dx * dx + dy * dy + dz * dz), 1e-12f);
      float bx = dx * inv, by = dy * inv, bz = dz * inv;

      float tx = ov[3 * i + 0], ty = ov[3 * i + 1], tz = ov[3 * i + 2];
      float dt = tx * bx + ty * by + tz * bz;
      float px = tx - dt * bx, py = ty - dt * by, pz = tz - dt * bz;
      float ip = 1.0f / fmaxf(sqrtf(px * px + py * py + pz * pz), 1e-12f);
      float ox = px * ip, oy = py * ip, oz = pz * ip;

      float zz = bx * oy - by * ox;                // cross(b,o).z (only z needed)
      float sy = sqrtf(bx * bx + by * by);
      res[8 + 3 * i + 0] = atan2f(oz, zz);         // ex
      res[8 + 3 * i + 1] = atan2f(-bz, sy);        // ey
      res[8 + 3 * i + 2] = atan2f(by, bx);         // ez
    }

    const float BLO[8] = {0.f, 0.f, -0.5f, -0.5f, 0.f, 0.f, -1.f, -1.f};
    const float BHI[8] = {2.f, 2.f, 0.5f, 0.5f, 2.6f, 2.6f, 1.f, 1.f};
#pragma unroll
    for (int k = 0; k < 8; ++k) {
      float s = sc[8 + k];
      res[38 + k] = (tanhf(s) + 1.0f) * 0.5f * (BHI[k] - BLO[k]) + BLO[k];
    }
  }
  __syncthreads();                                  // all LDS input reads done

  // ---- Stage output rows in LDS (stride 46 dwords: conflict-free for wave32),
  //      then drain as a linear coalesced b128 copy. ----
  float* outL = s_kp;                               // reuse input staging area
  if (t < nb) {
#pragma unroll
    for (int k = 0; k < 46; ++k) outL[t * 46 + k] = res[k];
  }
  __syncthreads();

  char* gdst = (char*)(out + (size_t)base * 46);    // base*184 B, 16B-aligned
  if (nb == TILE) {
    // TILE*46/4 = 2944 float4 chunks = 11 full sweeps + half sweep.
#pragma unroll
    for (int it = 0; it < 11; ++it) {
      int c = t + it * TILE;
      async_l2g_b128(gdst + (size_t)c * 16, (const char*)outL + (size_t)c * 16);
    }
    if (t < TILE / 2) {
      int c = t + 11 * TILE;
      async_l2g_b128(gdst + (size_t)c * 16, (const char*)outL + (size_t)c * 16);
    }
  } else {
    const int totalF = nb * 46;
    const int nvec = totalF >> 2;
    for (int c = t; c < nvec; c += TILE)
      async_l2g_b128(gdst + (size_t)c * 16, (const char*)outL + (size_t)c * 16);
    for (int k = (nvec << 2) + t; k < totalF; k += TILE)   // tail (nb odd only)
      out[(size_t)base * 46 + k] = outL[k];
  }
  wait_async_zero();   // S_ENDPGM also implies wait-idle; explicit for clarity
}

extern "C" void kernel_launch(void* const* d_in, const int* in_sizes, int n_in,
                              void* d_out, int out_size, void* d_ws, size_t ws_size,
                              hipStream_t stream) {
  const float* kp3d  = (const float*)d_in[0];   // (B, 24, 3)
  const float* ortho = (const float*)d_in[1];   // (B, 12, 3)
  const float* scal  = (const float*)d_in[2];   // (B, 16)
  float* out = (float*)d_out;                   // (B, 46)
  const int B = in_sizes[0] / 72;
  const int blocks = (B + TILE - 1) / TILE;
  bio_otsr_kernel<<<dim3(blocks), dim3(TILE), 0, stream>>>(kp3d, ortho, scal, out, B);
}